// NonlinearFilter_35081292873902
// MI455X (gfx1250) — compile-verified
//
#include <hip/hip_runtime.h>
#include <hip/hip_bf16.h>
#include <math.h>

// Problem dims (fixed by reference setup_inputs)
#define T_N 500
#define S_N 32
#define B_N 128
#define L_N 64
#define R_N 8

typedef __attribute__((ext_vector_type(2))) float v2f;
typedef __attribute__((ext_vector_type(8))) float v8f;

// ---------------------------------------------------------------------------
// CDNA5 f32 WMMA: D(16x16) = A(16x4) * B(4x16) + C   (full fp32 precision)
// ---------------------------------------------------------------------------
__device__ __forceinline__ v8f wmma4(v2f a, v2f b, v8f c) {
  return __builtin_amdgcn_wmma_f32_16x16x4_f32(
      /*neg_a=*/false, a, /*neg_b=*/false, b,
      /*c_mod=*/(short)0, c, /*reuse_a=*/false, /*reuse_b=*/false);
}

// 8x8 Cholesky in-place (lower triangle), serial on one thread (tiny).
__device__ __forceinline__ void chol8_t0(float* M) {
  for (int j = 0; j < 8; ++j) {
    float d = sqrtf(M[j * 8 + j]);
    M[j * 8 + j] = d;
    float di = 1.0f / d;
    for (int i = j + 1; i < 8; ++i) M[i * 8 + j] *= di;
    for (int c2 = j + 1; c2 < 8; ++c2)
      for (int i = c2; i < 8; ++i) M[i * 8 + c2] -= M[i * 8 + j] * M[c2 * 8 + j];
  }
}

// Solve (L L^T) x = v for the 8x8 factor, in place on v (per-thread, registers).
__device__ __forceinline__ void solve8_row(const float* Lm, float* v) {
  float y[8];
#pragma unroll
  for (int i = 0; i < 8; ++i) {
    float s = v[i];
#pragma unroll
    for (int k2 = 0; k2 < 8; ++k2)
      if (k2 < i) s -= Lm[i * 8 + k2] * y[k2];
    y[i] = s / Lm[i * 8 + i];
  }
#pragma unroll
  for (int i = 7; i >= 0; --i) {
    float s = y[i];
#pragma unroll
    for (int k2 = 7; k2 >= 0; --k2)
      if (k2 > i) s -= Lm[k2 * 8 + i] * v[k2];
    v[i] = s / Lm[i * 8 + i];
  }
}

// Solve (L L^T) x = v for the 32x32 factor; v in/out, y = LDS temp (thread 0).
__device__ void solve32_t0(const float* Lm, float* v, float* y) {
  for (int i = 0; i < 32; ++i) {
    float s = v[i];
    for (int k2 = 0; k2 < i; ++k2) s -= Lm[i * 32 + k2] * y[k2];
    y[i] = s / Lm[i * 32 + i];
  }
  for (int i = 31; i >= 0; --i) {
    float s = y[i];
    for (int k2 = i + 1; k2 < 32; ++k2) s -= Lm[k2 * 32 + i] * v[k2];
    v[i] = s / Lm[i * 32 + i];
  }
}

// ---------------------------------------------------------------------------
// Persistent kernel: one workgroup per batch b; loops over all T steps.
// block = 256 threads = 8 wave32s.
// ---------------------------------------------------------------------------
__global__ __launch_bounds__(256, 1) void nlf_persistent(
    const float* __restrict__ m0g, const float* __restrict__ q0g,
    const float* __restrict__ qg, const float* __restrict__ kg,
    const float* __restrict__ Kg, const float* __restrict__ Ag,
    const float* __restrict__ wp0g, const float* __restrict__ wf0g,
    const float* __restrict__ wp1g, const float* __restrict__ wp2g,
    const float* __restrict__ wfg, float* __restrict__ outg) {
  __shared__ float sA[L_N * L_N];    // A (row-major), resident all steps
  __shared__ float sZ[S_N * L_N];    // ensemble z [s][l]; reused as gz mid-step
  __shared__ float sMc[L_N * S_N];   // mthT then centered M_c [l][s]
  __shared__ float sZpc[S_N * L_N];  // z_p_c [s][l]
  __shared__ float sG[S_N * S_N];    // H = I+G, then its Cholesky L (lower)
  __shared__ float sWT[S_N * S_N];   // wp1 staging, later tmpS2 (Mc^T gz)
  __shared__ float sKt[L_N * R_N];   // K_t [l][q]
  __shared__ float sKM[R_N * S_N];   // K^T M_c
  __shared__ float sKPK[R_N * R_N];  // I + KPK, then its Cholesky (lower)
  __shared__ float sYs[S_N * R_N];   // v1t rows, then solved ys
  __shared__ float sQi[L_N], sQs[L_N], sQd[L_N];
  __shared__ float sMp[L_N], sH[L_N], sU1[L_N], sHp[L_N];
  __shared__ float sU[L_N], sMf[L_N], sGv[L_N];
  __shared__ float sT32a[S_N], sT32b[S_N], sT8[R_N];

  const int tid = threadIdx.x;
  const int b = blockIdx.x;
  const int lane = tid & 31;
  const int wv = tid >> 5;   // wave id 0..7
  const int lh = lane >> 4;  // lane-half (WMMA layout)
  const int l15 = lane & 15;
  const float cS = 0.17677669529663687f;  // 1/sqrt(S)

  // ---- constants into LDS ----
  for (int i = tid; i < L_N * L_N; i += 256) sA[i] = Ag[i];
  if (tid < L_N) {
    float qd = qg[tid];
    sQd[tid] = qd;
    sQi[tid] = 1.0f / qd;
    sQs[tid] = sqrtf(qd);
  }
  // ---- step 0 ----
  for (int i = tid; i < L_N * R_N; i += 256)
    sKt[i] = Kg[(size_t)b * L_N * R_N + i];  // K[0,b]
  if (tid < L_N) {
    float Pp = q0g[tid];
    sU1[tid] = Pp;                                      // stash Pp
    sGv[tid] = sqrtf(Pp);                               // stash sqrt(Pp)
    sH[tid] = m0g[tid] / Pp + kg[(size_t)b * L_N + tid];  // h = m0/Pp + k0
  }
  __syncthreads();
  for (int e = tid; e < S_N * L_N; e += 256) {  // z_p_c0 = sqrt(Pp)*w_p0
    int s = e >> 6, l = e & 63;
    sZpc[e] = sGv[l] * wp0g[((size_t)s * B_N + b) * L_N + l];
  }
  if (tid < R_N * R_N) {  // I + K0^T diag(Pp) K0
    int q = tid >> 3, p = tid & 7;
    float a = (q == p) ? 1.0f : 0.0f;
    for (int l = 0; l < L_N; ++l) a += sKt[l * R_N + q] * sU1[l] * sKt[l * R_N + p];
    sKPK[tid] = a;
  }
  __syncthreads();
  if (tid == 0) chol8_t0(sKPK);
  __syncthreads();
  if (tid < R_N) {  // (P_p K)^T h
    float a = 0.f;
    for (int l = 0; l < L_N; ++l) a += sU1[l] * sKt[l * R_N + tid] * sH[l];
    sT8[tid] = a;
  }
  __syncthreads();
  if (tid == 0) solve8_row(sKPK, sT8);
  __syncthreads();
  if (tid < L_N) {  // m0f = Pp*(h - K0 y)
    float a = 0.f;
    for (int q = 0; q < R_N; ++q) a += sKt[tid * R_N + q] * sT8[q];
    float mf = sU1[tid] * (sH[tid] - a);
    sMf[tid] = mf;
    outg[(size_t)b * L_N + tid] = mf;  // out[0,b,:]
  }
  __syncthreads();
  {  // v1 = K0^T z_p_c + w_f0  (256 threads = 32x8 exactly)
    int s = tid >> 3, q = tid & 7;
    float a = wf0g[((size_t)s * B_N + b) * R_N + q];
    for (int l = 0; l < L_N; ++l) a += sKt[l * R_N + q] * sZpc[s * L_N + l];
    sYs[s * R_N + q] = a;
  }
  __syncthreads();
  if (tid < S_N) solve8_row(sKPK, &sYs[tid * R_N]);
  __syncthreads();
  for (int e = tid; e < S_N * L_N; e += 256) {  // z0
    int s = e >> 6, l = e & 63;
    float a = 0.f;
    for (int q = 0; q < R_N; ++q) a += sKt[l * R_N + q] * sYs[s * R_N + q];
    sZ[e] = sMf[l] + sZpc[e] - sU1[l] * a;
  }
  __syncthreads();

  // ---- steps 1..T-1 (fully resident scan) ----
  for (int t = 1; t < T_N; ++t) {
    const size_t tb = (size_t)(t - 1);
    // stage wp1 tile and K_t; prefetch next step's stream into caches
    for (int i = tid; i < S_N * S_N; i += 256) {
      int s = i >> 5, ii = i & 31;
      sWT[i] = wp1g[((tb * S_N + s) * B_N + b) * S_N + ii];
    }
    for (int i = tid; i < L_N * R_N; i += 256)
      sKt[i] = Kg[((size_t)t * B_N + b) * L_N * R_N + i];
    if (t + 1 < T_N) {
      if (tid < S_N) {
        __builtin_prefetch(&wp2g[(((size_t)t * S_N + tid) * B_N + b) * L_N], 0, 1);
        __builtin_prefetch(&wp1g[(((size_t)t * S_N + tid) * B_N + b) * S_N], 0, 1);
        __builtin_prefetch(&wfg[(((size_t)t * S_N + tid) * B_N + b) * R_N], 0, 1);
      }
      if (tid < 16)
        __builtin_prefetch(&Kg[((size_t)(t + 1) * B_N + b) * L_N * R_N + tid * 32], 0, 1);
      if (tid == 0)
        __builtin_prefetch(&kg[((size_t)(t + 1) * B_N + b) * L_N], 0, 1);
    }
    __syncthreads();

    // (a) mthT = z + 0.1*tanh(Z @ A^T), stored transposed [l][s]  (WMMA f32)
    {
      int m0 = (wv >> 2) << 4, n0 = (wv & 3) << 4;  // 2x4 tiles, 8 waves
      v8f c = {0.f, 0.f, 0.f, 0.f, 0.f, 0.f, 0.f, 0.f};
#pragma unroll
      for (int k4 = 0; k4 < L_N; k4 += 4) {
        int kb = k4 + lh * 2;
        v2f a, bb;
        a.x = sZ[(m0 + l15) * L_N + kb];
        a.y = sZ[(m0 + l15) * L_N + kb + 1];
        bb.x = sA[(n0 + l15) * L_N + kb];  // B(k,n) = A[n][k]  (A^T)
        bb.y = sA[(n0 + l15) * L_N + kb + 1];
        c = wmma4(a, bb, c);
      }
#pragma unroll
      for (int v = 0; v < 8; ++v) {
        int row = m0 + v + lh * 8, col = n0 + l15;  // (s,l)
        sMc[col * S_N + row] = sZ[row * L_N + col] + 0.1f * tanhf(c[v]);
      }
    }
    __syncthreads();

    // mean over S, center & scale -> M_c
    if (tid < L_N) {
      float m = 0.f;
#pragma unroll
      for (int s = 0; s < S_N; ++s) m += sMc[tid * S_N + s];
      sMp[tid] = m * (1.0f / S_N);
    }
    __syncthreads();
    for (int e = tid; e < L_N * S_N; e += 256) {
      int l = e >> 5;
      sMc[e] = cS * (sMc[e] - sMp[l]);
    }
    __syncthreads();

    // (b) H = I + Mc^T diag(Qi) Mc   (WMMA, 4 tiles on waves 0..3)
    if (wv < 4) {
      int m0 = (wv >> 1) << 4, n0 = (wv & 1) << 4;
      v8f c = {0.f, 0.f, 0.f, 0.f, 0.f, 0.f, 0.f, 0.f};
#pragma unroll
      for (int k4 = 0; k4 < L_N; k4 += 4) {
        int kb = k4 + lh * 2;
        v2f a, bb;
        a.x = sMc[kb * S_N + m0 + l15] * sQi[kb];
        a.y = sMc[(kb + 1) * S_N + m0 + l15] * sQi[kb + 1];
        bb.x = sMc[kb * S_N + n0 + l15];
        bb.y = sMc[(kb + 1) * S_N + n0 + l15];
        c = wmma4(a, bb, c);
      }
#pragma unroll
      for (int v = 0; v < 8; ++v) {
        int row = m0 + v + lh * 8, col = n0 + l15;
        sG[row * S_N + col] = c[v] + ((row == col) ? 1.0f : 0.0f);
      }
    }
    __syncthreads();

    // cooperative 32x32 Cholesky of H (right-looking, LDS)
    for (int j = 0; j < S_N; ++j) {
      if (tid == 0) sG[j * S_N + j] = sqrtf(sG[j * S_N + j]);
      __syncthreads();
      if (tid > j && tid < S_N) sG[tid * S_N + j] /= sG[j * S_N + j];
      __syncthreads();
      for (int e = tid; e < S_N * S_N; e += 256) {
        int i = e >> 5, c2 = e & 31;
        if (c2 > j && i >= c2) sG[e] -= sG[i * S_N + j] * sG[c2 * S_N + j];
      }
      __syncthreads();
    }

    // h_p = u1 - Qi * Mc (H^-1 (Mc^T u1)),  h = h_p + k_t
    if (tid < L_N) sU1[tid] = sQi[tid] * sMp[tid];
    __syncthreads();
    if (tid < S_N) {
      float a = 0.f;
#pragma unroll
      for (int l = 0; l < L_N; ++l) a += sMc[l * S_N + tid] * sU1[l];
      sT32a[tid] = a;
    }
    __syncthreads();
    if (tid == 0) solve32_t0(sG, sT32a, sT32b);
    __syncthreads();
    if (tid < L_N) {
      float a = 0.f;
#pragma unroll
      for (int i = 0; i < S_N; ++i) a += sMc[tid * S_N + i] * sT32a[i];
      sHp[tid] = sU1[tid] - sQi[tid] * a;
      sH[tid] = sHp[tid] + kg[((size_t)t * B_N + b) * L_N + tid];
    }
    __syncthreads();

    // (c) z_p_c = Wp1 @ Mc^T + Qs*wp2   (WMMA, 8 tiles)
    {
      int m0 = (wv >> 2) << 4, n0 = (wv & 3) << 4;
      v8f c = {0.f, 0.f, 0.f, 0.f, 0.f, 0.f, 0.f, 0.f};
#pragma unroll
      for (int k4 = 0; k4 < S_N; k4 += 4) {
        int kb = k4 + lh * 2;
        v2f a, bb;
        a.x = sWT[(m0 + l15) * S_N + kb];
        a.y = sWT[(m0 + l15) * S_N + kb + 1];
        bb.x = sMc[(n0 + l15) * S_N + kb];  // B(k=i,n=l) = Mc[l][i]
        bb.y = sMc[(n0 + l15) * S_N + kb + 1];
        c = wmma4(a, bb, c);
      }
#pragma unroll
      for (int v = 0; v < 8; ++v) {
        int row = m0 + v + lh * 8, col = n0 + l15;  // (s,l)
        float w2 = wp2g[((tb * S_N + row) * B_N + b) * L_N + col];
        sZpc[row * L_N + col] = c[v] + sQs[col] * w2;
      }
    }
    __syncthreads();

    // KM = K^T Mc  (exactly 8 waves x 32 lanes)
    {
      int q = wv, i = lane;
      float a = 0.f;
#pragma unroll
      for (int l = 0; l < L_N; ++l) a += sKt[l * R_N + q] * sMc[l * S_N + i];
      sKM[q * S_N + i] = a;
    }
    __syncthreads();
    // I + KM KM^T + K^T diag(Q) K
    if (tid < R_N * R_N) {
      int q = tid >> 3, p = tid & 7;
      float a = (q == p) ? 1.0f : 0.0f;
#pragma unroll
      for (int i = 0; i < S_N; ++i) a += sKM[q * S_N + i] * sKM[p * S_N + i];
#pragma unroll
      for (int l = 0; l < L_N; ++l) a += sKt[l * R_N + q] * sQd[l] * sKt[l * R_N + p];
      sKPK[tid] = a;
    }
    __syncthreads();
    if (tid == 0) chol8_t0(sKPK);
    __syncthreads();

    // u = P_p_mv(h)
    if (tid < S_N) {
      float a = 0.f;
#pragma unroll
      for (int l = 0; l < L_N; ++l) a += sMc[l * S_N + tid] * sH[l];
      sT32a[tid] = a;
    }
    __syncthreads();
    if (tid < L_N) {
      float a = 0.f;
#pragma unroll
      for (int i = 0; i < S_N; ++i) a += sMc[tid * S_N + i] * sT32a[i];
      sU[tid] = a + sQd[tid] * sH[tid];
    }
    __syncthreads();
    // y = (I+KPK)^-1 (K^T u);  g = K y
    if (tid < R_N) {
      float a = 0.f;
#pragma unroll
      for (int l = 0; l < L_N; ++l) a += sKt[l * R_N + tid] * sU[l];
      sT8[tid] = a;
    }
    __syncthreads();
    if (tid == 0) solve8_row(sKPK, sT8);
    __syncthreads();
    if (tid < L_N) {
      float a = 0.f;
#pragma unroll
      for (int q = 0; q < R_N; ++q) a += sKt[tid * R_N + q] * sT8[q];
      sGv[tid] = a;
    }
    __syncthreads();
    // m_f = u - P_p_mv(g); write output row
    if (tid < S_N) {
      float a = 0.f;
#pragma unroll
      for (int l = 0; l < L_N; ++l) a += sMc[l * S_N + tid] * sGv[l];
      sT32a[tid] = a;
    }
    __syncthreads();
    if (tid < L_N) {
      float a = 0.f;
#pragma unroll
      for (int i = 0; i < S_N; ++i) a += sMc[tid * S_N + i] * sT32a[i];
      float mf = sU[tid] - (a + sQd[tid] * sGv[tid]);
      sMf[tid] = mf;
      outg[((size_t)t * B_N + b) * L_N + tid] = mf;
    }
    __syncthreads();

    // v1t = K^T z_p_c + wf; solve per ensemble member
    {
      int s = tid >> 3, q = tid & 7;
      float a = wfg[((tb * S_N + s) * B_N + b) * R_N + q];
#pragma unroll
      for (int l = 0; l < L_N; ++l) a += sKt[l * R_N + q] * sZpc[s * L_N + l];
      sYs[s * R_N + q] = a;
    }
    __syncthreads();
    if (tid < S_N) solve8_row(sKPK, &sYs[tid * R_N]);
    __syncthreads();
    // gz = K ys  (reuse sZ; ensemble z is dead after phase (a))
    for (int e = tid; e < S_N * L_N; e += 256) {
      int s = e >> 6, l = e & 63;
      float a = 0.f;
#pragma unroll
      for (int q = 0; q < R_N; ++q) a += sKt[l * R_N + q] * sYs[s * R_N + q];
      sZ[e] = a;
    }
    __syncthreads();
    // (d) tmpS2 = Gz @ Mc   (WMMA, 4 tiles)
    if (wv < 4) {
      int m0 = (wv >> 1) << 4, n0 = (wv & 1) << 4;
      v8f c = {0.f, 0.f, 0.f, 0.f, 0.f, 0.f, 0.f, 0.f};
#pragma unroll
      for (int k4 = 0; k4 < L_N; k4 += 4) {
        int kb = k4 + lh * 2;
        v2f a, bb;
        a.x = sZ[(m0 + l15) * L_N + kb];
        a.y = sZ[(m0 + l15) * L_N + kb + 1];
        bb.x = sMc[kb * S_N + n0 + l15];
        bb.y = sMc[(kb + 1) * S_N + n0 + l15];
        c = wmma4(a, bb, c);
      }
#pragma unroll
      for (int v = 0; v < 8; ++v)
        sWT[(m0 + v + lh * 8) * S_N + n0 + l15] = c[v];
    }
    __syncthreads();
    // (e) Pz = tmpS2 @ Mc^T; z_new = m_f + z_p_c - (Pz + Qd*gz)  -> sZ
    {
      int m0 = (wv >> 2) << 4, n0 = (wv & 3) << 4;
      v8f c = {0.f, 0.f, 0.f, 0.f, 0.f, 0.f, 0.f, 0.f};
#pragma unroll
      for (int k4 = 0; k4 < S_N; k4 += 4) {
        int kb = k4 + lh * 2;
        v2f a, bb;
        a.x = sWT[(m0 + l15) * S_N + kb];
        a.y = sWT[(m0 + l15) * S_N + kb + 1];
        bb.x = sMc[(n0 + l15) * S_N + kb];
        bb.y = sMc[(n0 + l15) * S_N + kb + 1];
        c = wmma4(a, bb, c);
      }
#pragma unroll
      for (int v = 0; v < 8; ++v) {
        int row = m0 + v + lh * 8, col = n0 + l15;  // (s,l), disjoint per lane
        float gzv = sZ[row * L_N + col];
        sZ[row * L_N + col] =
            sMf[col] + sZpc[row * L_N + col] - (c[v] + sQd[col] * gzv);
      }
    }
    __syncthreads();
  }
}

// ---------------------------------------------------------------------------
extern "C" void kernel_launch(void* const* d_in, const int* in_sizes, int n_in,
                              void* d_out, int out_size, void* d_ws,
                              size_t ws_size, hipStream_t stream) {
  (void)in_sizes; (void)n_in; (void)out_size; (void)d_ws; (void)ws_size;
  const float* m0 = (const float*)d_in[0];
  const float* q0 = (const float*)d_in[1];
  const float* qd = (const float*)d_in[2];
  const float* k = (const float*)d_in[3];
  const float* K = (const float*)d_in[4];
  const float* A = (const float*)d_in[5];
  const float* wp0 = (const float*)d_in[6];
  const float* wf0 = (const float*)d_in[7];
  const float* wp1 = (const float*)d_in[8];
  const float* wp2 = (const float*)d_in[9];
  const float* wf = (const float*)d_in[10];
  float* out = (float*)d_out;
  nlf_persistent<<<dim3(B_N), dim3(256), 0, stream>>>(
      m0, q0, qd, k, K, A, wp0, wf0, wp1, wp2, wf, out);
}